// ParallelExperts_5978594476260
// MI455X (gfx1250) — compile-verified
//
#include <hip/hip_runtime.h>
#include <hip/hip_bf16.h>

// ---------------------------------------------------------------------------
// MoE scatter2scatter (top-2, 8 experts, 1024x1024 fp32) for gfx1250.
// Kernel 1: LDS-tiled grouped GEMM over expert-sorted rows using
//           V_WMMA_F32_16X16X4_F32. B tiles staged via
//           GLOBAL_LOAD_ASYNC_TO_LDS_B128 (ASYNCcnt), A tiles gathered +
//           expert-masked through registers, ds_store_b128.
// Kernel 2: gate-weighted combine out[n] = g0*of[2n] + g1*of[2n+1].
// ---------------------------------------------------------------------------

typedef __attribute__((ext_vector_type(2))) float v2f;
typedef __attribute__((ext_vector_type(8))) float v8f;

namespace {
constexpr int D_IN     = 1024;
constexpr int D_OUT    = 1024;
constexpr int TOP_K    = 2;
constexpr int N_TOKENS = 4096;
constexpr int N_ROWS   = N_TOKENS * TOP_K;   // 8192 routed rows

constexpr int BM = 128;          // sorted rows per workgroup tile
constexpr int BN = 128;          // output cols per workgroup tile
constexpr int BK = 16;           // k-depth per stage
constexpr int A_STRIDE = 20;     // floats; padded vs 16 to kill bank conflicts
constexpr int B_STRIDE = 136;    // floats; padded vs 128 (lane16-31 bank shift)
constexpr int N_STAGES = D_IN / BK;   // 64
}

// Async global->LDS copy of 16 bytes (per-lane addresses), tracked by ASYNCcnt.
__device__ __forceinline__ void async_copy_b128(const float* g, float* l) {
    const unsigned int  la = (unsigned int)(unsigned long long)l;   // LDS offset = addr[31:0]
    const unsigned long long ga = (unsigned long long)l ? (unsigned long long)g
                                                        : (unsigned long long)g; // keep types simple
    asm volatile("global_load_async_to_lds_b128 %0, %1, off"
                 :
                 : "v"(la), "v"(ga)
                 : "memory");
}

__device__ __forceinline__ void async_wait0() {
    asm volatile("s_wait_asynccnt 0x0" ::: "memory");
}

// ---------------------------------------------------------------------------
// Grouped GEMM: workgroup tile 128 (sorted rows) x 128 (cols), 8 wave32.
// Wave (wm, wn) in 2x4 grid owns a 64x32 slab: 4 M-subtiles x 2 N-subtiles.
// ---------------------------------------------------------------------------
__global__ __launch_bounds__(256) void moe_wmma_gemm(
    const float* __restrict__ x,      // [N_TOKENS, D_IN]
    const float* __restrict__ W,      // [E, D_IN, D_OUT]
    const int*   __restrict__ sexp,   // [N_ROWS] sorted expert ids (ascending)
    const int*   __restrict__ sscat,  // [N_ROWS] flat slot index per sorted row
    float*       __restrict__ outflat // [N_ROWS, D_OUT] flat-slot order
) {
    __shared__ float Abuf[2][BM * A_STRIDE];   // 2 x 10240 B
    __shared__ float Bbuf[2][BK * B_STRIDE];   // 2 x  8704 B

    const int tid    = threadIdx.x;
    const int lane   = tid & 31;
    const int wave   = tid >> 5;
    const int waveM  = wave >> 2;              // 0..1
    const int waveN  = wave & 3;               // 0..3
    const int lane15 = lane & 15;
    const int koff   = (lane >> 4) << 1;       // 0 / 2 (A,B K-split across half-waves)

    const int m0 = blockIdx.x * BM;
    const int n0 = blockIdx.y * BN;

    // ---- A staging assignment: thread -> (row, 8-float segment) ----
    const int arow = tid >> 1;                 // 0..127
    const int aseg = tid & 1;                  // 0..1
    const int tokA = sscat[m0 + arow] / TOP_K; // gather source token
    const int expA = sexp[m0 + arow];          // this row's expert
    const float* xA = x + (size_t)tokA * D_IN + aseg * 8;

    // ---- B staging assignment: thread -> two float4 of the 16x128 tile ----
    const int brow0 = tid >> 5;                // 0..7   (+8 for second)
    const int bcol4 = tid & 31;                // 0..31

    const int eLo = sexp[m0];
    const int eHi = sexp[m0 + BM - 1];

    v8f acc[4][2];
    #pragma unroll
    for (int s = 0; s < 4; ++s)
        #pragma unroll
        for (int t = 0; t < 2; ++t)
            acc[s][t] = (v8f){};

    for (int e = eLo; e <= eHi; ++e) {
        const float maskA = (expA == e) ? 1.0f : 0.0f;
        const float* __restrict__ Wp = W + (size_t)e * ((size_t)D_IN * D_OUT);

        // ---- stage helpers ----
        auto stageB = [&](int k0, int buf) {
            #pragma unroll
            for (int i = 0; i < 2; ++i) {
                const int row = brow0 + i * 8;
                const float* g = Wp + (size_t)(k0 + row) * D_OUT + n0 + bcol4 * 4;
                float* l = &Bbuf[buf][row * B_STRIDE + bcol4 * 4];
                async_copy_b128(g, l);
            }
        };
        auto stageA = [&](int k0, int buf) {
            float4 p0 = *(const float4*)(xA + k0);
            float4 p1 = *(const float4*)(xA + k0 + 4);
            p0.x *= maskA; p0.y *= maskA; p0.z *= maskA; p0.w *= maskA;
            p1.x *= maskA; p1.y *= maskA; p1.z *= maskA; p1.w *= maskA;
            float* dst = &Abuf[buf][arow * A_STRIDE + aseg * 8];
            *(float4*)(dst)     = p0;
            *(float4*)(dst + 4) = p1;
        };

        // ---- prologue: stage k0 = 0 into buffer 0 ----
        stageB(0, 0);
        stageA(0, 0);
        async_wait0();
        __syncthreads();

        // ---- main pipeline ----
        for (int s = 0; s < N_STAGES; ++s) {
            const int cur = s & 1;
            const int nxt = cur ^ 1;
            const int k0  = s * BK;

            if (s + 1 < N_STAGES) {            // prefetch next stage
                stageB(k0 + BK, nxt);
                stageA(k0 + BK, nxt);
            }

            // compute on current buffers
            #pragma unroll
            for (int kk = 0; kk < BK; kk += 4) {
                v2f a[4];
                #pragma unroll
                for (int sm = 0; sm < 4; ++sm) {
                    const int row = waveM * 64 + sm * 16 + lane15;
                    a[sm] = *(const v2f*)&Abuf[cur][row * A_STRIDE + kk + koff];
                }
                v2f b[2];
                #pragma unroll
                for (int nt = 0; nt < 2; ++nt) {
                    const int col = waveN * 32 + nt * 16 + lane15;
                    b[nt].x = Bbuf[cur][(kk + koff)     * B_STRIDE + col];
                    b[nt].y = Bbuf[cur][(kk + koff + 1) * B_STRIDE + col];
                }
                #pragma unroll
                for (int sm = 0; sm < 4; ++sm)
                    #pragma unroll
                    for (int nt = 0; nt < 2; ++nt)
                        acc[sm][nt] = __builtin_amdgcn_wmma_f32_16x16x4_f32(
                            false, a[sm], false, b[nt], (short)0,
                            acc[sm][nt], false, false);
            }

            async_wait0();                     // async B prefetch landed in LDS
            __syncthreads();                   // (compiler adds dscnt wait for A)
        }
    }

    // ---- epilogue: scatter D to flat-slot order ----
    // D layout: vgpr r -> M = r + 8*(lane>=16), N = lane&15.
    const int mhalf = (lane >> 4) << 3;
    #pragma unroll
    for (int sm = 0; sm < 4; ++sm) {
        #pragma unroll
        for (int r = 0; r < 8; ++r) {
            const int m  = m0 + waveM * 64 + sm * 16 + r + mhalf;
            const int jj = sscat[m];
            float* dst = outflat + (size_t)jj * D_OUT + n0 + waveN * 32 + lane15;
            dst[0]  = acc[sm][0][r];
            dst[16] = acc[sm][1][r];
        }
    }
}

// ---------------------------------------------------------------------------
// Gate-weighted combine: out[n,:] = g[n,0]*of[2n,:] + g[n,1]*of[2n+1,:]
// ---------------------------------------------------------------------------
__global__ __launch_bounds__(256) void moe_combine(
    const float* __restrict__ outflat,  // [N_ROWS, D_OUT]
    const float* __restrict__ gates,    // [N_TOKENS, TOP_K]
    float*       __restrict__ out       // [N_TOKENS, D_OUT]
) {
    constexpr int D4 = D_OUT / 4;
    const int i  = blockIdx.x * blockDim.x + threadIdx.x;   // over N_TOKENS*D4
    const int n  = i / D4;
    const int d4 = i - n * D4;

    const float g0 = gates[n * TOP_K + 0];
    const float g1 = gates[n * TOP_K + 1];

    const float4 a = ((const float4*)(outflat + (size_t)(TOP_K * n)     * D_OUT))[d4];
    const float4 b = ((const float4*)(outflat + (size_t)(TOP_K * n + 1) * D_OUT))[d4];

    float4 r;
    r.x = g0 * a.x + g1 * b.x;
    r.y = g0 * a.y + g1 * b.y;
    r.z = g0 * a.z + g1 * b.z;
    r.w = g0 * a.w + g1 * b.w;
    ((float4*)(out + (size_t)n * D_OUT))[d4] = r;
}

extern "C" void kernel_launch(void* const* d_in, const int* in_sizes, int n_in,
                              void* d_out, int out_size, void* d_ws, size_t ws_size,
                              hipStream_t stream) {
    const float* x     = (const float*)d_in[0];   // [4096,1024] f32
    const float* W     = (const float*)d_in[1];   // [8,1024,1024] f32
    const float* gates = (const float*)d_in[2];   // [4096,2] f32
    // d_in[3] = k (scalar, ==2), d_in[6]/d_in[7] bookkeeping: unused in math
    const int* sexp  = (const int*)d_in[4];       // [8192] sorted expert ids
    const int* sscat = (const int*)d_in[5];       // [8192] flat slot indices

    float* outflat = (float*)d_ws;                // 8192*1024*4 = 32 MB scratch
    float* out     = (float*)d_out;

    dim3 grid_g(N_ROWS / BM, D_OUT / BN);         // 64 x 8
    moe_wmma_gemm<<<grid_g, dim3(256), 0, stream>>>(x, W, sexp, sscat, outflat);

    const int total4 = N_TOKENS * (D_OUT / 4);
    moe_combine<<<dim3(total4 / 256), dim3(256), 0, stream>>>(outflat, gates, out);
}